// Network_15461882266003
// MI455X (gfx1250) — compile-verified
//
#include <hip/hip_runtime.h>

// ---------------------------------------------------------------------------
// Persistent-kernel recurrent network simulation for MI455X (gfx1250).
// One kernel runs all 100 timesteps; one device-wide barrier per step.
// GEMMs use V_WMMA_F32_16X16X4_F32 (exact fp32 matrix pipe).
// ---------------------------------------------------------------------------

typedef float v2f __attribute__((ext_vector_type(2)));
typedef float v8f __attribute__((ext_vector_type(8)));

#define WMMA_F32(a, b, c) \
  __builtin_amdgcn_wmma_f32_16x16x4_f32(false, (a), false, (b), (short)0, (c), false, false)

constexpr int kB  = 32;     // batch
constexpr int kT  = 100;    // timesteps
constexpr int kN  = 2048;   // neurons
constexpr int kA  = 1536;   // NA0 (STP population)
constexpr int kWin = 5;     // output windows

constexpr int kBlocks  = kN / 16;   // 128 blocks, one 16-col tile each
constexpr int kThreads = 128;       // 4 waves: 4-way K split

__device__ __forceinline__ void grid_sync(unsigned int* bar, unsigned int nblk) {
  __threadfence();            // release: flush this thread's global writes
  __syncthreads();
  if (threadIdx.x == 0) {
    unsigned int ticket = atomicAdd(bar, 1u);
    unsigned int target = (ticket / nblk + 1u) * nblk;   // monotonic generations
    while (__hip_atomic_load(bar, __ATOMIC_RELAXED, __HIP_MEMORY_SCOPE_AGENT) < target) {
      __builtin_amdgcn_s_sleep(1);
    }
  }
  __syncthreads();
  __threadfence();            // acquire: invalidate stale WGP$ lines
}

__global__ __launch_bounds__(kThreads)
void net_persistent(const float* __restrict__ ff,        // (32,100,2048)
                    const float* __restrict__ Wab,       // (2048,2048) k-major
                    const float* __restrict__ Wstp,      // (1536,1536) k-major
                    const float* __restrict__ rec_init,  // (32,2048)
                    float* __restrict__ out,             // (32,5,1536)
                    float* __restrict__ rates2,          // 2 * 32*2048 (ping-pong)
                    float* __restrict__ rec,             // 32*2048
                    float* __restrict__ u2,              // 2 * 32*1536
                    float* __restrict__ x2,              // 2 * 32*1536
                    float* __restrict__ aux2,            // 2 * 32*1536
                    unsigned int* __restrict__ bar) {
  const unsigned int G = gridDim.x;
  const int tid  = threadIdx.x;
  const int wave = tid >> 5;
  const int lane = tid & 31;
  const int lo   = lane & 15;     // column within tile / A row
  const int hi   = lane >> 4;     // half-wave select (K pair / C row half)
  const int col0 = blockIdx.x * 16;

  const float E_TAU = 0.9512294245f;   // exp(-DT/TAU),     DT/TAU     = 0.05
  const float E_SYN = 0.7788007831f;   // exp(-DT/TAU_SYN), DT/TAU_SYN = 0.25

  __shared__ float slab[4][512];       // per-wave partial 32x16 tiles (8 KB)

  // ---------------- init: rates0 = relu(ff[:,0] + rec_init), u=USE, x=1 ----
  for (int e = tid; e < 512; e += kThreads) {
    const int b = e >> 4, c = e & 15, n = col0 + c;
    const float fi  = ff[(b * kT + 0) * kN + n];
    const float rin = rec_init[b * kN + n];
    const float r0  = fmaxf(fi + rin, 0.0f);
    rates2[b * kN + n] = r0;           // buffer 0 = "cur" at t=0
    rec[b * kN + n]    = rin;
    if (n < kA) {
      u2[b * kA + n]   = 0.03f;
      x2[b * kA + n]   = 1.0f;
      aux2[b * kA + n] = 0.03f * r0;   // u0*x0*re0
      for (int w = 0; w < kWin; ++w) out[(b * kWin + w) * kA + n] = 0.0f;
    }
  }
  grid_sync(bar, G);

  // ---------------- time loop --------------------------------------------
  for (int t = 0; t < kT; ++t) {
    const int cur = t & 1, nxt = cur ^ 1;
    const float* rates_c = rates2 + cur * (kB * kN);
    float*       rates_n = rates2 + nxt * (kB * kN);
    const float* u_c   = u2   + cur * (kB * kA);
    float*       u_n   = u2   + nxt * (kB * kA);
    const float* x_c   = x2   + cur * (kB * kA);
    float*       x_n   = x2   + nxt * (kB * kA);
    const float* aux_c = aux2 + cur * (kB * kA);
    float*       aux_n = aux2 + nxt * (kB * kA);

    v8f c0a = {}, c0b = {}, c1a = {}, c1b = {};   // rows 0-15 / 16-31, 2-way ILP

    // ---- hidden += rates @ Wab_T : this wave's K quarter (512 of 2048) ----
    {
      const int kbeg = wave * (kN / 4), kend = kbeg + (kN / 4);
      #pragma unroll 2
      for (int k0 = kbeg; k0 < kend; k0 += 8) {
        const int kk = k0 + 2 * hi;
        v2f a0 = *(const v2f*)&rates_c[lo * kN + kk];
        v2f a1 = *(const v2f*)&rates_c[(16 + lo) * kN + kk];
        v2f bb; bb.x = Wab[kk * kN + col0 + lo];
                bb.y = Wab[(kk + 1) * kN + col0 + lo];
        c0a = WMMA_F32(a0, bb, c0a);
        c1a = WMMA_F32(a1, bb, c1a);
        const int k2 = kk + 4;
        v2f a2 = *(const v2f*)&rates_c[lo * kN + k2];
        v2f a3 = *(const v2f*)&rates_c[(16 + lo) * kN + k2];
        v2f b2; b2.x = Wab[k2 * kN + col0 + lo];
                b2.y = Wab[(k2 + 1) * kN + col0 + lo];
        c0b = WMMA_F32(a2, b2, c0b);
        c1b = WMMA_F32(a3, b2, c1b);
      }
    }
    // ---- hidden[:, :1536] += (u*x*rates) @ W_stp_T  (J_STP == 1.0) --------
    if (col0 < kA) {
      const int kbeg = wave * (kA / 4), kend = kbeg + (kA / 4);
      #pragma unroll 2
      for (int k0 = kbeg; k0 < kend; k0 += 8) {
        const int kk = k0 + 2 * hi;
        v2f a0 = *(const v2f*)&aux_c[lo * kA + kk];
        v2f a1 = *(const v2f*)&aux_c[(16 + lo) * kA + kk];
        v2f bb; bb.x = Wstp[kk * kA + col0 + lo];
                bb.y = Wstp[(kk + 1) * kA + col0 + lo];
        c0a = WMMA_F32(a0, bb, c0a);
        c1a = WMMA_F32(a1, bb, c1a);
        const int k2 = kk + 4;
        v2f a2 = *(const v2f*)&aux_c[lo * kA + k2];
        v2f a3 = *(const v2f*)&aux_c[(16 + lo) * kA + k2];
        v2f b2; b2.x = Wstp[k2 * kA + col0 + lo];
                b2.y = Wstp[(k2 + 1) * kA + col0 + lo];
        c0b = WMMA_F32(a2, b2, c0b);
        c1b = WMMA_F32(a3, b2, c1b);
      }
    }

    // ---- deposit partial tile (C/D layout: VGPR v -> row v + 8*hi) --------
    #pragma unroll
    for (int v = 0; v < 8; ++v) {
      slab[wave][(v + 8 * hi) * 16 + lo]      = c0a[v] + c0b[v];
      slab[wave][(16 + v + 8 * hi) * 16 + lo] = c1a[v] + c1b[v];
    }
    __syncthreads();

    // ---- deterministic K reduction + elementwise dynamics (owner tile) ----
    for (int e = tid; e < 512; e += kThreads) {
      const float h = slab[0][e] + slab[1][e] + slab[2][e] + slab[3][e];
      const int b = e >> 4, c = e & 15, n = col0 + c;
      float r = rec[b * kN + n];
      r = r * E_SYN + h * 0.25f;                    // rec update
      rec[b * kN + n] = r;
      const float net = ff[(b * kT + t) * kN + n] + r;
      const float re  = rates_c[b * kN + n];        // pre-update rates
      const float rt  = re * E_TAU + fmaxf(net, 0.0f) * 0.05f;
      rates_n[b * kN + n] = rt;
      if (n < kA) {
        const float uo = u_c[b * kA + n];
        const float xo = x_c[b * kA + n];
        const float un = uo + (0.03f - uo) / 1000.0f + 0.03f * (1.0f - uo) * re;
        const float xn = xo + (1.0f - xo) / 250.0f - un * xo * re;  // uses NEW u
        u_n[b * kA + n]   = un;
        x_n[b * kA + n]   = xn;
        aux_n[b * kA + n] = un * xn * rt;            // aux for step t+1
        int win = -1;
        if (t >= 40 && t <= 50)       win = 0;                 // 11-step window
        else if (t >= 51 && t <= 90)  win = 1 + (t - 51) / 10; // 10-step windows
        if (win >= 0) out[(b * kWin + win) * kA + n] += rt * 0.1f;
      }
    }

    grid_sync(bar, G);   // also guards slab[] reuse next iteration
  }
}

// ---------------------------------------------------------------------------
// Workspace layout (floats):
//   [0, 131072)        rates ping-pong (2 * 32*2048)
//   [131072, 196608)   rec
//   [196608, 294912)   u ping-pong
//   [294912, 393216)   x ping-pong
//   [393216, 491520)   aux ping-pong
//   [491520]           barrier counter (unsigned int)
// ---------------------------------------------------------------------------
extern "C" void kernel_launch(void* const* d_in, const int* in_sizes, int n_in,
                              void* d_out, int out_size, void* d_ws, size_t ws_size,
                              hipStream_t stream) {
  const float* ff       = (const float*)d_in[0];  // (32,100,2048)
  const float* Wab      = (const float*)d_in[1];  // (2048,2048)
  const float* Wstp     = (const float*)d_in[2];  // (1536,1536)
  const float* rec_init = (const float*)d_in[3];  // (32,2048)
  float* out = (float*)d_out;                     // (32,5,1536)

  float* ws     = (float*)d_ws;
  float* rates2 = ws;
  float* rec    = ws + 131072;
  float* u2     = ws + 196608;
  float* x2     = ws + 294912;
  float* aux2   = ws + 393216;
  unsigned int* bar = (unsigned int*)(ws + 491520);

  hipMemsetAsync(bar, 0, sizeof(unsigned int), stream);  // fresh barrier each call
  net_persistent<<<kBlocks, kThreads, 0, stream>>>(
      ff, Wab, Wstp, rec_init, out, rates2, rec, u2, x2, aux2, bar);
}